// ADSR_59889023976186
// MI455X (gfx1250) — compile-verified
//
#include <hip/hip_runtime.h>
#include <stdint.h>
#include <math.h>

// ADSR envelope = four chained associative scans over T samples.
// Memory-bound (64 MiB I/O, ~3us HBM floor on MI455X; all intermediate traffic
// L2-resident in the 192MB L2). No GEMM structure -> WMMA not applicable;
// CDNA5 path used: async global->LDS staging (global_load_async_to_lds_b128 +
// s_wait_asynccnt) + wave32 LDS hierarchical scans. Inter-block combine
// kernels are single-workgroup *parallel* scans (no serial 1-thread tails).

#define CHUNK 8192
#define BLOCK 256
#define SEG (CHUNK / BLOCK) // 32 samples per thread

// ---------------------------------------------------------------- staging ---
__device__ __forceinline__ void stage_gate(const float* __restrict__ gate,
                                           long long base, int T,
                                           float* s_gate) {
  const int tid = threadIdx.x;
  if (base + CHUNK <= (long long)T) {
    // Each thread async-copies its 32 floats (128B) as 8x B128 into LDS.
    const float* src = gate + base + (long long)tid * SEG;
    unsigned lds = (unsigned)(uintptr_t)(&s_gate[tid * SEG]);
#pragma unroll
    for (int k = 0; k < SEG / 4; ++k) {
      asm volatile("global_load_async_to_lds_b128 %0, %1, off"
                   :
                   : "v"(lds + 16u * (unsigned)k), "v"(src + 4 * k)
                   : "memory");
    }
    asm volatile("s_wait_asynccnt 0" ::: "memory");
  } else {
    for (int j = tid; j < CHUNK; j += BLOCK) {
      long long gi = base + j;
      s_gate[j] = (gi < (long long)T) ? gate[gi] : 0.f;
    }
  }
  __syncthreads();
}

// ------------------------------------------------------------- LDS scans ----
// Run-of-ones scan element: (c = trailing-ones count, flags bit0 = has_zero,
// bit1 = any_gate).  combine(P,C): c = C.z ? C.c : P.c + C.c ; flags |=.
__device__ __forceinline__ void scan_run(float* sc, unsigned* sf) {
  const int tid = threadIdx.x;
  for (int off = 1; off < BLOCK; off <<= 1) {
    float c = sc[tid];
    unsigned f = sf[tid];
    float cp = 0.f;
    unsigned fp = 0u;
    if (tid >= off) { cp = sc[tid - off]; fp = sf[tid - off]; }
    __syncthreads();
    if (tid >= off) {
      if (!(f & 1u)) c = cp + c;
      f |= fp;
    }
    sc[tid] = c; sf[tid] = f;
    __syncthreads();
  }
}

// Segmented-sum ("tail since last reset") scan: (t, r = has_reset).
// combine(P,C): C.r ? (C.t, 1) : (P.t + C.t, P.r)
__device__ __forceinline__ void scan_tail(float* st, unsigned* sr) {
  const int tid = threadIdx.x;
  for (int off = 1; off < BLOCK; off <<= 1) {
    float t = st[tid];
    unsigned r = sr[tid];
    float tp = 0.f;
    unsigned rp = 0u;
    if (tid >= off) { tp = st[tid - off]; rp = sr[tid - off]; }
    __syncthreads();
    if (tid >= off) {
      if (r == 0u) { t = tp + t; r = rp; }
    }
    st[tid] = t; sr[tid] = r;
    __syncthreads();
  }
}

// Plain float sum scan (inclusive).
__device__ __forceinline__ void scan_sum(float* ss) {
  const int tid = threadIdx.x;
  for (int off = 1; off < BLOCK; off <<= 1) {
    float t = ss[tid];
    float tp = (tid >= off) ? ss[tid - off] : 0.f;
    __syncthreads();
    ss[tid] = t + tp;
    __syncthreads();
  }
}

// ------------------------------------------------------------------- K1a ----
// Per-block (trailing-ones count, has_zero, any_gate).
__global__ __launch_bounds__(BLOCK) void k1a(const float* __restrict__ gate,
                                             int T, float* bs_c, float* bs_z,
                                             float* bs_ag) {
  __shared__ float s_gate[CHUNK];
  __shared__ float sc[BLOCK];
  __shared__ unsigned sf[BLOCK];
  const int tid = threadIdx.x;
  const long long base = (long long)blockIdx.x * CHUNK;
  stage_gate(gate, base, T, s_gate);

  const int lbase = tid * SEG;
  float c = 0.f;
  bool z = false, ag = false;
  for (int j = 0; j < SEG; ++j) {
    if (base + lbase + j >= (long long)T) break;
    float g = s_gate[lbase + j];
    if (g != 0.f) { c += 1.f; ag = true; } else { c = 0.f; z = true; }
  }
  sc[tid] = c;
  sf[tid] = (z ? 1u : 0u) | (ag ? 2u : 0u);
  __syncthreads();
  scan_run(sc, sf);
  if (tid == BLOCK - 1) {
    int b = blockIdx.x;
    bs_c[b] = sc[tid];
    bs_z[b] = (sf[tid] & 1u) ? 1.f : 0.f;
    bs_ag[b] = (sf[tid] & 2u) ? 1.f : 0.f;
  }
}

// ------------------------------------------------------------------- K1b ----
// Parallel single-workgroup combine: per-block exclusive (n_in, seen_in).
// With initial state (n=0), the exclusive running count equals the scanned
// aggregate count directly.
__global__ __launch_bounds__(BLOCK) void k1b(int NB, const float* bs_c,
                                             const float* bs_z,
                                             const float* bs_ag, float* in_n,
                                             float* in_seen) {
  __shared__ float sc[BLOCK];
  __shared__ unsigned sf[BLOCK];
  const int tid = threadIdx.x;
  const int PER = (NB + BLOCK - 1) / BLOCK;
  const int e0 = tid * PER;

  float c = 0.f;
  unsigned f = 0u; // identity: no zero, no gate
  for (int k = 0; k < PER; ++k) {
    int e = e0 + k;
    if (e >= NB) break;
    float ec = bs_c[e];
    unsigned ez = (bs_z[e] != 0.f) ? 1u : 0u;
    unsigned eag = (bs_ag[e] != 0.f) ? 2u : 0u;
    if (!ez) ec = c + ec;
    c = ec;
    f = f | ez | eag;
  }
  sc[tid] = c;
  sf[tid] = f;
  __syncthreads();
  scan_run(sc, sf);

  float n = 0.f;
  bool seen = false;
  if (tid > 0) {
    n = sc[tid - 1]; // running value from 0-init == aggregate count
    seen = (sf[tid - 1] & 2u) != 0u;
  }
  for (int k = 0; k < PER; ++k) {
    int e = e0 + k;
    if (e >= NB) break;
    in_n[e] = n;
    in_seen[e] = seen ? 1.f : 0.f;
    float ec = bs_c[e];
    n = (bs_z[e] != 0.f) ? ec : n + ec;
    seen = seen || (bs_ag[e] != 0.f);
  }
}

// ------------------------------------------------------------------- K1c ----
// Exact ad_axis / log_rs, local segmented cum_log scan, attack-seed summary.
__global__ __launch_bounds__(BLOCK) void k1c(
    const float* __restrict__ gate, int T, const float* in_n,
    const float* in_seen, const float* pA, const float* pD, const float* pS,
    const float* pR, float* cl_tail, float* cl_r, float* att_s, float* att_h,
    float* att_d) {
  __shared__ float s_gate[CHUNK];
  __shared__ float s_logrs[CHUNK];
  __shared__ float sc[BLOCK];
  __shared__ unsigned sf[BLOCK];
  __shared__ float st[BLOCK];
  __shared__ unsigned sr[BLOCK];
  __shared__ float ssum[BLOCK];
  __shared__ float sh_d;
  __shared__ unsigned sh_h;

  const int tid = threadIdx.x;
  const int b = blockIdx.x;
  const long long base = (long long)b * CHUNK;
  const float attack = *pA, decay = *pD, sustain = *pS, release = *pR;
  const float rtc = expf(-1.f / release);
  const float inv_decay = 1.f / decay;
  const bool att_ok = (1.f <= attack);

  if (tid == 0) { sh_h = 0u; sh_d = 0.f; }
  stage_gate(gate, base, T, s_gate);

  const int lbase = tid * SEG;
  // ---- sweep 1: run / any-gate scan -> per-thread incoming (n, seen)
  {
    float c = 0.f;
    bool z = false, ag = false;
    for (int j = 0; j < SEG; ++j) {
      if (base + lbase + j >= (long long)T) break;
      float g = s_gate[lbase + j];
      if (g != 0.f) { c += 1.f; ag = true; } else { c = 0.f; z = true; }
    }
    sc[tid] = c;
    sf[tid] = (z ? 1u : 0u) | (ag ? 2u : 0u);
  }
  __syncthreads();
  scan_run(sc, sf);
  float n_in_t;
  bool seen_in_t;
  {
    float nin = in_n[b];
    bool sin = in_seen[b] != 0.f;
    if (tid == 0) { n_in_t = nin; seen_in_t = sin; }
    else {
      float c = sc[tid - 1];
      unsigned f = sf[tid - 1];
      n_in_t = (f & 1u) ? c : nin + c;
      seen_in_t = sin || ((f & 2u) != 0u);
    }
  }
  __syncthreads();

  // ---- sweep 2: log_rs + segment (tail, has_reset)
  float gp = (tid == 0) ? ((base > 0) ? gate[base - 1] : 0.f)
                        : s_gate[lbase - 1];
  {
    float n = n_in_t;
    bool seen = seen_in_t;
    float t = 0.f;
    bool r = false;
    for (int j = 0; j < SEG; ++j) {
      if (base + lbase + j >= (long long)T) break;
      float g = s_gate[lbase + j];
      float lr;
      if (g != 0.f) {
        lr = 0.f;
      } else if (!seen) {
        lr = 0.f; // ri == 0 before first note -> log_rs = 0
      } else {
        float sv = 1.f;
        if (gp != 0.f && n > attack)
          sv = sustain + (1.f - sustain) * expf(-(n - attack) * inv_decay);
        float ri = sv * rtc;
        lr = (ri > 0.f) ? logf(ri) : 0.f;
      }
      s_logrs[lbase + j] = lr;
      if (lr == 0.f) { t = 0.f; r = true; } else { t += lr; }
      n = (g != 0.f) ? n + 1.f : 0.f;
      seen = seen || (g != 0.f);
      gp = g;
    }
    st[tid] = t;
    sr[tid] = r ? 1u : 0u;
  }
  __syncthreads();
  scan_tail(st, sr);
  float p_in = (tid == 0) ? 0.f : st[tid - 1];
  bool rb = (tid == 0) ? false : (sr[tid - 1] != 0u);

  // ---- sweep 3: attack seeds (local sum + unique C_in-dependent (h,d))
  gp = (tid == 0) ? ((base > 0) ? gate[base - 1] : 0.f) : s_gate[lbase - 1];
  {
    float p = p_in;
    bool reset = rb;
    float acc = 0.f;
    for (int j = 0; j < SEG; ++j) {
      if (base + lbase + j >= (long long)T) break;
      float g = s_gate[lbase + j];
      if (gp == 0.f && g != 0.f && att_ok) {
        if (!reset) { sh_h = 1u; sh_d = p; } // unique per block
        else acc += (p == 0.f) ? 0.f : expf(p);
      }
      float lr = s_logrs[lbase + j];
      if (lr == 0.f) { p = 0.f; reset = true; } else { p += lr; }
      gp = g;
    }
    ssum[tid] = acc;
  }
  __syncthreads();
  scan_sum(ssum);
  if (tid == BLOCK - 1) {
    cl_tail[b] = st[tid];
    cl_r[b] = (sr[tid] != 0u) ? 1.f : 0.f;
    att_s[b] = ssum[tid];
  }
  __syncthreads();
  if (tid == 0) {
    att_h[b] = sh_h ? 1.f : 0.f;
    att_d[b] = sh_d;
  }
}

// -------------------------------------------------------------------- K2 ----
// Parallel single-workgroup combine: exclusive tail-scan gives C_in per block;
// then the nonlinear seed exp(C_in + d) is pointwise and A_in is an exclusive
// plain-sum scan.
__global__ __launch_bounds__(BLOCK) void k2(int NB, const float* cl_tail,
                                            const float* cl_r,
                                            const float* att_s,
                                            const float* att_h,
                                            const float* att_d, float* C_in,
                                            float* A_in) {
  __shared__ float st[BLOCK];
  __shared__ unsigned sr[BLOCK];
  __shared__ float ss[BLOCK];
  const int tid = threadIdx.x;
  const int PER = (NB + BLOCK - 1) / BLOCK;
  const int e0 = tid * PER;

  // phase 1: per-thread tail aggregate over its entries
  float t = 0.f;
  unsigned r = 0u;
  for (int k = 0; k < PER; ++k) {
    int e = e0 + k;
    if (e >= NB) break;
    float et = cl_tail[e];
    if (cl_r[e] != 0.f) { t = et; r = 1u; } else { t = t + et; }
  }
  st[tid] = t;
  sr[tid] = r;
  __syncthreads();
  scan_tail(st, sr);
  float C = (tid == 0) ? 0.f : st[tid - 1]; // 0-init => running value = tail
  // write C_in, accumulate per-thread sigma partial
  float sig = 0.f;
  for (int k = 0; k < PER; ++k) {
    int e = e0 + k;
    if (e >= NB) break;
    C_in[e] = C;
    float s = att_s[e];
    if (att_h[e] != 0.f) {
      float c = C + att_d[e];
      s += (c == 0.f) ? 0.f : expf(c);
    }
    sig += s;
    float et = cl_tail[e];
    C = (cl_r[e] != 0.f) ? et : C + et;
  }
  ss[tid] = sig;
  __syncthreads();
  scan_sum(ss);
  float A = (tid == 0) ? 0.f : ss[tid - 1];
  for (int k = 0; k < PER; ++k) {
    int e = e0 + k;
    if (e >= NB) break;
    A_in[e] = A;
    float s = att_s[e];
    if (att_h[e] != 0.f) {
      float c = C_in[e] + att_d[e]; // same-thread reload, program-ordered
      s += (c == 0.f) ? 0.f : expf(c);
    }
    A += s;
  }
}

// -------------------------------------------------------------------- K3 ----
__global__ __launch_bounds__(BLOCK) void k3(
    const float* __restrict__ gate, int T, const float* in_n,
    const float* in_seen, const float* C_in_a, const float* A_in_a,
    const float* pA, const float* pD, const float* pS, const float* pR,
    float* __restrict__ out) {
  __shared__ float s_gate[CHUNK];
  __shared__ float s_logrs[CHUNK];
  __shared__ float sc[BLOCK];
  __shared__ unsigned sf[BLOCK];
  __shared__ float st[BLOCK];
  __shared__ unsigned sr[BLOCK];
  __shared__ float ssum[BLOCK];

  const int tid = threadIdx.x;
  const int b = blockIdx.x;
  const long long base = (long long)b * CHUNK;
  const float attack = *pA, decay = *pD, sustain = *pS, release = *pR;
  const float rtc = expf(-1.f / release);
  const float inv_decay = 1.f / decay;
  const bool att_ok = (1.f <= attack);
  const float Cin = C_in_a[b];
  const float Ain = A_in_a[b];

  stage_gate(gate, base, T, s_gate);
  const int lbase = tid * SEG;

  // ---- sweep 1: run scan
  {
    float c = 0.f;
    bool z = false, ag = false;
    for (int j = 0; j < SEG; ++j) {
      if (base + lbase + j >= (long long)T) break;
      float g = s_gate[lbase + j];
      if (g != 0.f) { c += 1.f; ag = true; } else { c = 0.f; z = true; }
    }
    sc[tid] = c;
    sf[tid] = (z ? 1u : 0u) | (ag ? 2u : 0u);
  }
  __syncthreads();
  scan_run(sc, sf);
  float n_in_t;
  bool seen_in_t;
  {
    float nin = in_n[b];
    bool sin = in_seen[b] != 0.f;
    if (tid == 0) { n_in_t = nin; seen_in_t = sin; }
    else {
      float c = sc[tid - 1];
      unsigned f = sf[tid - 1];
      n_in_t = (f & 1u) ? c : nin + c;
      seen_in_t = sin || ((f & 2u) != 0u);
    }
  }
  __syncthreads();

  // ---- sweep 2: log_rs + cum_log segment scan
  float gp0 = (tid == 0) ? ((base > 0) ? gate[base - 1] : 0.f)
                         : s_gate[lbase - 1];
  {
    float gp = gp0, n = n_in_t;
    bool seen = seen_in_t;
    float t = 0.f;
    bool r = false;
    for (int j = 0; j < SEG; ++j) {
      if (base + lbase + j >= (long long)T) break;
      float g = s_gate[lbase + j];
      float lr;
      if (g != 0.f) lr = 0.f;
      else if (!seen) lr = 0.f;
      else {
        float sv = 1.f;
        if (gp != 0.f && n > attack)
          sv = sustain + (1.f - sustain) * expf(-(n - attack) * inv_decay);
        float ri = sv * rtc;
        lr = (ri > 0.f) ? logf(ri) : 0.f;
      }
      s_logrs[lbase + j] = lr;
      if (lr == 0.f) { t = 0.f; r = true; } else { t += lr; }
      n = (g != 0.f) ? n + 1.f : 0.f;
      seen = seen || (g != 0.f);
      gp = g;
    }
    st[tid] = t;
    sr[tid] = r ? 1u : 0u;
  }
  __syncthreads();
  scan_tail(st, sr);
  float p_in = (tid == 0) ? 0.f : st[tid - 1];
  bool rb = (tid == 0) ? false : (sr[tid - 1] != 0u);

  // ---- sweep 3: attack-seed segment sums (exact, Cin known)
  {
    float gp = gp0, p = p_in;
    bool reset = rb;
    float acc = 0.f;
    for (int j = 0; j < SEG; ++j) {
      if (base + lbase + j >= (long long)T) break;
      float g = s_gate[lbase + j];
      if (gp == 0.f && g != 0.f && att_ok) {
        float c = reset ? p : Cin + p; // cum_log[i-1]
        acc += (c == 0.f) ? 0.f : expf(c);
      }
      float lr = s_logrs[lbase + j];
      if (lr == 0.f) { p = 0.f; reset = true; } else { p += lr; }
      gp = g;
    }
    ssum[tid] = acc;
  }
  __syncthreads();
  scan_sum(ssum);
  float a = Ain + ((tid == 0) ? 0.f : ssum[tid - 1]);

  // ---- sweep 4: final outputs
  {
    float gp = gp0, n = n_in_t, p = p_in;
    bool reset = rb;
    for (int j = 0; j < SEG; ++j) {
      long long idx = base + lbase + j;
      if (idx >= (long long)T) break;
      float g = s_gate[lbase + j];
      if (gp == 0.f && g != 0.f && att_ok) {
        float c = reset ? p : Cin + p; // cum_log[i-1]
        a += (c == 0.f) ? 0.f : expf(c);
      }
      float lr = s_logrs[lbase + j];
      bool nr = (lr == 0.f);
      float pn = nr ? 0.f : p + lr;
      bool rn = reset || nr;
      float o;
      if (g != 0.f) {
        float nn = n + 1.f;
        if (nn <= attack) o = (1.f - a) * (nn / attack) + a;
        else o = sustain + (1.f - sustain) * expf(-(nn - attack) * inv_decay);
        n = nn;
      } else {
        float c = rn ? pn : Cin + pn; // cum_log[i]
        o = (c == 0.f) ? 0.f : expf(c);
        n = 0.f;
      }
      out[idx] = o;
      p = pn;
      reset = rn;
      gp = g;
    }
  }
}

// ---------------------------------------------------------------- launch ----
extern "C" void kernel_launch(void* const* d_in, const int* in_sizes, int n_in,
                              void* d_out, int out_size, void* d_ws,
                              size_t ws_size, hipStream_t stream) {
  const float* gate = (const float*)d_in[0];
  const float* pA = (const float*)d_in[1];
  const float* pD = (const float*)d_in[2];
  const float* pS = (const float*)d_in[3];
  const float* pR = (const float*)d_in[4];
  float* out = (float*)d_out;
  const int T = in_sizes[0];
  if (T <= 0) return;
  const int NB = (T + CHUNK - 1) / CHUNK;

  float* ws = (float*)d_ws;
  float* bs_c = ws + (size_t)0 * NB;
  float* bs_z = ws + (size_t)1 * NB;
  float* bs_ag = ws + (size_t)2 * NB;
  float* in_n = ws + (size_t)3 * NB;
  float* in_seen = ws + (size_t)4 * NB;
  float* cl_tail = ws + (size_t)5 * NB;
  float* cl_r = ws + (size_t)6 * NB;
  float* att_s = ws + (size_t)7 * NB;
  float* att_h = ws + (size_t)8 * NB;
  float* att_d = ws + (size_t)9 * NB;
  float* C_in = ws + (size_t)10 * NB;
  float* A_in = ws + (size_t)11 * NB;

  k1a<<<NB, BLOCK, 0, stream>>>(gate, T, bs_c, bs_z, bs_ag);
  k1b<<<1, BLOCK, 0, stream>>>(NB, bs_c, bs_z, bs_ag, in_n, in_seen);
  k1c<<<NB, BLOCK, 0, stream>>>(gate, T, in_n, in_seen, pA, pD, pS, pR,
                                cl_tail, cl_r, att_s, att_h, att_d);
  k2<<<1, BLOCK, 0, stream>>>(NB, cl_tail, cl_r, att_s, att_h, att_d, C_in,
                              A_in);
  k3<<<NB, BLOCK, 0, stream>>>(gate, T, in_n, in_seen, C_in, A_in, pA, pD, pS,
                               pR, out);
}